// FourEightSTEQuantizer_22471268893171
// MI455X (gfx1250) — compile-verified
//
#include <hip/hip_runtime.h>
#include <stdint.h>

// ---------------------------------------------------------------------------
// FourEightSTEQuantizer forward:
//   per-row (4096) RMS scale -> 4-bit mid-rise quantize -> per-8-group 2:4
//   pair sparsification (zero the 2 smallest-L2 pairs of 4, norms on raw x).
// Memory-bound: 256 MiB traffic -> ~11us floor @ 23.3 TB/s.
// Data path: TDM (tensor_load_to_lds) stages each 16KB row into LDS,
// s_wait_tensorcnt + workgroup barrier, conflict-free ds_load_b128 compute,
// non-temporal global_store_b128 (write-once stream, > L2 capacity).
// ---------------------------------------------------------------------------

typedef unsigned int v4u __attribute__((ext_vector_type(4)));
typedef int          v4i __attribute__((ext_vector_type(4)));
typedef int          v8i __attribute__((ext_vector_type(8)));
typedef float        v4f __attribute__((ext_vector_type(4)));

#define ROW_LEN   4096
#define THREADS   256
#define F4_ROW    (ROW_LEN / 4)        // 1024 float4 per row
#define F4_THREAD (F4_ROW / THREADS)   // 4 float4 per thread

#if defined(__HIP_DEVICE_COMPILE__)
#if __has_builtin(__builtin_amdgcn_tensor_load_to_lds) && \
    __has_builtin(__builtin_amdgcn_s_wait_tensorcnt)
#define USE_TDM 1
#endif
#endif

#define GAUSS_SCALE_4BIT 2.513930578568423f

__global__ __launch_bounds__(THREADS)
void FourEightSTEQuantizer_22471268893171_kernel(const float* __restrict__ x,
                                                 float* __restrict__ out) {
#ifdef USE_TDM
  __shared__ float4 s_row[F4_ROW];          // 16 KB staging tile
#endif
  __shared__ float s_red[THREADS / 32];
  __shared__ float s_scale;

  const int t = threadIdx.x;
  const long long row = blockIdx.x;
  const float* __restrict__ xrow = x + row * (long long)ROW_LEN;

  float4 f[F4_THREAD];

#ifdef USE_TDM
  // ---- wave 0 issues the TDM load of the whole row into LDS -------------
  // readfirstlane makes the predicate wave-uniform so the compiler emits a
  // scalar branch: TDM ignores EXEC, so waves 1..7 must *branch over* it.
  const int wave = __builtin_amdgcn_readfirstlane((int)(threadIdx.x >> 5));
  if (wave == 0) {
    const unsigned long long ga = (unsigned long long)(uintptr_t)xrow;
    const unsigned lds_base = (unsigned)(uintptr_t)(void*)&s_row[0];

    // D# group 0 (128b): count=1 | lds_addr | global_addr[56:0] | type=2
    v4u g0;
    g0.x = 1u;                                         // count=1, user mode
    g0.y = lds_base;                                   // LDS byte address
    g0.z = (unsigned)(ga & 0xFFFFFFFFull);             // global_addr[31:0]
    g0.w = ((unsigned)((ga >> 32) & 0x1FFFFFFull))     // global_addr[56:32]
           | 0x80000000u;                              // type=2 ("image")
    // D# group 1 (256b): 2D tensor, data_size=4B, dim0=4096, 1 row tile
    v8i g1;
    g1[0] = (int)(2u << 16);        // workgroup_mask=0, data_size=2 (4B)
    g1[1] = (int)(4096u << 16);     // tensor_dim0[15:0] in [31:16]
    g1[2] = (int)(1u << 16);        // tensor_dim0[31:16]=0, tensor_dim1=1
    g1[3] = (int)(4096u << 16);     // tensor_dim1[31:16]=0, tile_dim0=4096
    g1[4] = 1;                      // tile_dim1=1, tile_dim2=0
    g1[5] = 4096;                   // tensor_dim0_stride[31:0]
    g1[6] = 0;                      // stride0[47:32], stride1[15:0]
    g1[7] = 0;                      // stride1[47:16]
    v4i g2 = {0, 0, 0, 0};
    v4i g3 = {0, 0, 0, 0};
#if defined(__clang_major__) && (__clang_major__ >= 23)
    v8i g4 = {0, 0, 0, 0, 0, 0, 0, 0};
    __builtin_amdgcn_tensor_load_to_lds(g0, g1, g2, g3, g4, 0);
#else
    __builtin_amdgcn_tensor_load_to_lds(g0, g1, g2, g3, 0);
#endif
    __builtin_amdgcn_s_wait_tensorcnt(0);
  }
  __syncthreads();
#pragma unroll
  for (int i = 0; i < F4_THREAD; ++i)
    f[i] = s_row[i * THREADS + t];   // lane-contiguous 16B -> no bank conflicts
#else
  // Fallback: direct coalesced 128-bit global loads into registers.
  const float4* __restrict__ xrow4 = (const float4*)xrow;
#pragma unroll
  for (int i = 0; i < F4_THREAD; ++i)
    f[i] = xrow4[i * THREADS + t];
#endif

  // ---- row sum of squares -> RMS scale ----------------------------------
  float ss = 0.0f;
#pragma unroll
  for (int i = 0; i < F4_THREAD; ++i) {
    ss += f[i].x * f[i].x + f[i].y * f[i].y;
    ss += f[i].z * f[i].z + f[i].w * f[i].w;
  }
#pragma unroll
  for (int off = 16; off > 0; off >>= 1)
    ss += __shfl_down(ss, off, 32);
  if ((t & 31) == 0) s_red[t >> 5] = ss;
  __syncthreads();
  if (t == 0) {
    float tot = 0.0f;
#pragma unroll
    for (int j = 0; j < THREADS / 32; ++j) tot += s_red[j];
    s_scale = GAUSS_SCALE_4BIT * sqrtf(tot * (1.0f / (float)ROW_LEN)) + 1e-8f;
  }
  __syncthreads();
  const float scale = s_scale;
  const float step  = scale * (2.0f / 15.0f);   // 2*scale/(N_LEVELS-1)
  const float rstep = 1.0f / step;
  const float hstep = 0.5f * step;

  // ---- quantize + 2-of-4 pair sparsify, store ---------------------------
  v4f* __restrict__ orow4 = (v4f*)(out + row * (long long)ROW_LEN);
  const int base = (t & 1) << 1;   // this lane's global pair indices: base, base+1

#pragma unroll
  for (int i = 0; i < F4_THREAD; ++i) {
    const float4 v = f[i];
    // pair L2^2 on raw x (monotonic in the reference's L2 norm)
    const float p0 = v.x * v.x + v.y * v.y;
    const float p1 = v.z * v.z + v.w * v.w;
    const float q0 = __shfl_xor(p0, 1, 32);   // partner lane's pair sums
    const float q1 = __shfl_xor(p1, 1, 32);
    float sn[4];
    if (t & 1) { sn[0] = q0; sn[1] = q1; sn[2] = p0; sn[3] = p1; }
    else       { sn[0] = p0; sn[1] = p1; sn[2] = q0; sn[3] = q1; }
    // two smallest, first-occurrence ties (matches stable top_k(-norms,2))
    int i1 = 0;
    if (sn[1] < sn[i1]) i1 = 1;
    if (sn[2] < sn[i1]) i1 = 2;
    if (sn[3] < sn[i1]) i1 = 3;
    int i2 = -1;
#pragma unroll
    for (int j = 0; j < 4; ++j) {
      if (j == i1) continue;
      if (i2 < 0 || sn[j] < sn[i2]) i2 = j;
    }
    const float keep0 = (base == i1 || base == i2) ? 0.0f : 1.0f;
    const float keep1 = ((base + 1) == i1 || (base + 1) == i2) ? 0.0f : 1.0f;

    // mid-rise 4-bit quantizer, RNE rounding like jnp.round
    v4f o;
    {
      float c;
      c   = fminf(fmaxf(v.x, -scale), scale);
      o.x = (rintf(c * rstep + 0.5f) * step - hstep) * keep0;
      c   = fminf(fmaxf(v.y, -scale), scale);
      o.y = (rintf(c * rstep + 0.5f) * step - hstep) * keep0;
      c   = fminf(fmaxf(v.z, -scale), scale);
      o.z = (rintf(c * rstep + 0.5f) * step - hstep) * keep1;
      c   = fminf(fmaxf(v.w, -scale), scale);
      o.w = (rintf(c * rstep + 0.5f) * step - hstep) * keep1;
    }
    // write-once stream > L2 capacity: non-temporal 128-bit store
    __builtin_nontemporal_store(o, orow4 + (i * THREADS + t));
  }
}

extern "C" void kernel_launch(void* const* d_in, const int* in_sizes, int n_in,
                              void* d_out, int out_size, void* d_ws, size_t ws_size,
                              hipStream_t stream) {
  const float* x = (const float*)d_in[0];
  float* out = (float*)d_out;
  const int rows = in_sizes[0] / ROW_LEN;   // 8192
  FourEightSTEQuantizer_22471268893171_kernel<<<rows, THREADS, 0, stream>>>(x, out);
}